// Noise_Cell_37228776522108
// MI455X (gfx1250) — compile-verified
//
#include <hip/hip_runtime.h>

// out[i] = G[idx[i]] * (1 + 0.03 * eps[i])
// HBM-bound streaming op: 12 B per element, ~2 FLOPs. Optimal shape on MI455X:
//  - B128 non-temporal loads/stores (805 MB footprint > 192 MB L2 -> stream)
//  - 8-entry LUT staged in LDS, conflict-free ds_load_b32 gathers
//  - fast path with 8 unguarded B128 loads per thread for max MLP
//  - 32-bit chunk indices so loads use SADDR + 32-bit voffset (scale_offset)

#define NOISE_PARAM 0.03f
#define TPB 256
#define V4_PER_THREAD 4

typedef float  f32x4 __attribute__((ext_vector_type(4)));
typedef int    i32x4 __attribute__((ext_vector_type(4)));

__device__ __forceinline__ float noise_apply(float g, float e) {
    return g * __fmaf_rn(NOISE_PARAM, e, 1.0f);
}

__global__ __launch_bounds__(TPB) void noise_cell_kernel(
    const float* __restrict__ G,
    const float* __restrict__ eps,
    const int*   __restrict__ idx,
    float*       __restrict__ out,
    unsigned nv4,            // number of float4 chunks (fits in 32 bits)
    unsigned n)              // total elements (fits in 32 bits)
{
    // Stage the 8-entry conductance LUT into LDS.
    __shared__ float sG[8];
    if (threadIdx.x < 8) sG[threadIdx.x] = G[threadIdx.x];
    __syncthreads();

    // Block owns a contiguous span of TPB*V4_PER_THREAD chunks; thread t
    // touches chunks base + k*TPB (coalesced within each wave32).
    unsigned base = blockIdx.x * (TPB * V4_PER_THREAD) + threadIdx.x;

    if (base + (V4_PER_THREAD - 1) * TPB < nv4) {
        // ---- Fast path: whole span in range, no per-chunk predication. ----
        i32x4 vi[V4_PER_THREAD];
        f32x4 ve[V4_PER_THREAD];
#pragma unroll
        for (int k = 0; k < V4_PER_THREAD; ++k) {
            unsigned c = base + k * TPB;
            vi[k] = __builtin_nontemporal_load((const i32x4*)idx + c);
            ve[k] = __builtin_nontemporal_load((const f32x4*)eps + c);
        }
#pragma unroll
        for (int k = 0; k < V4_PER_THREAD; ++k) {
            unsigned c = base + k * TPB;
            f32x4 vo;
            vo.x = noise_apply(sG[vi[k].x & 7], ve[k].x);
            vo.y = noise_apply(sG[vi[k].y & 7], ve[k].y);
            vo.z = noise_apply(sG[vi[k].z & 7], ve[k].z);
            vo.w = noise_apply(sG[vi[k].w & 7], ve[k].w);
            __builtin_nontemporal_store(vo, (f32x4*)out + c);
        }
    } else {
        // ---- Guarded fallback for the (possibly partial) last block. ----
#pragma unroll
        for (int k = 0; k < V4_PER_THREAD; ++k) {
            unsigned c = base + k * TPB;
            if (c < nv4) {
                i32x4 vi = __builtin_nontemporal_load((const i32x4*)idx + c);
                f32x4 ve = __builtin_nontemporal_load((const f32x4*)eps + c);
                f32x4 vo;
                vo.x = noise_apply(sG[vi.x & 7], ve.x);
                vo.y = noise_apply(sG[vi.y & 7], ve.y);
                vo.z = noise_apply(sG[vi.z & 7], ve.z);
                vo.w = noise_apply(sG[vi.w & 7], ve.w);
                __builtin_nontemporal_store(vo, (f32x4*)out + c);
            }
        }
        // Tail elements (n % 4): first few threads of the last block.
        if (blockIdx.x == gridDim.x - 1) {
            unsigned t = nv4 * 4u + threadIdx.x;
            if (t < n) {
                out[t] = noise_apply(sG[idx[t] & 7], eps[t]);
            }
        }
    }
}

extern "C" void kernel_launch(void* const* d_in, const int* in_sizes, int n_in,
                              void* d_out, int out_size, void* d_ws, size_t ws_size,
                              hipStream_t stream) {
    // setup_inputs() order: G (8 x f32), eps (f32), idx (i32)
    const float* G   = (const float*)d_in[0];
    const float* eps = (const float*)d_in[1];
    const int*   idx = (const int*)d_in[2];
    float*       out = (float*)d_out;

    unsigned n   = (unsigned)out_size;   // 67,108,864 for this workload
    unsigned nv4 = n / 4u;

    unsigned per_block = TPB * V4_PER_THREAD;
    unsigned blocks = (nv4 + per_block - 1) / per_block;
    if (blocks == 0) blocks = 1;

    noise_cell_kernel<<<blocks, TPB, 0, stream>>>(G, eps, idx, out, nv4, n);
}